// LC_LOSS_80951543595379
// MI455X (gfx1250) — compile-verified
//
#include <hip/hip_runtime.h>

typedef __attribute__((ext_vector_type(2))) float v2f;
typedef __attribute__((ext_vector_type(8))) float v8f;

#define TPB 64
#define PRED_F 90
#define Y_F 99
#define GRID_A 2048

// ---------- small fp32 helpers (all in registers) ----------

// Zhou et al. 6D -> 3x3 rotation, columns = (b1, b2, b1 x b2). Row-major R[r*3+c].
__device__ __forceinline__ void gs6(const float* __restrict__ p, float R[9]) {
    float a1x = p[0], a1y = p[1], a1z = p[2];
    float a2x = p[3], a2y = p[4], a2z = p[5];
    float n1 = sqrtf(a1x*a1x + a1y*a1y + a1z*a1z);
    float i1 = 1.0f / fmaxf(n1, 1e-12f);
    float b1x = a1x*i1, b1y = a1y*i1, b1z = a1z*i1;
    float d  = b1x*a2x + b1y*a2y + b1z*a2z;
    float ux = a2x - d*b1x, uy = a2y - d*b1y, uz = a2z - d*b1z;
    float n2 = sqrtf(ux*ux + uy*uy + uz*uz);
    float i2 = 1.0f / fmaxf(n2, 1e-12f);
    float b2x = ux*i2, b2y = uy*i2, b2z = uz*i2;
    float b3x = b1y*b2z - b1z*b2y;
    float b3y = b1z*b2x - b1x*b2z;
    float b3z = b1x*b2y - b1y*b2x;
    R[0]=b1x; R[1]=b2x; R[2]=b3x;
    R[3]=b1y; R[4]=b2y; R[5]=b3y;
    R[6]=b1z; R[7]=b2z; R[8]=b3z;
}

__device__ __forceinline__ void mm3(const float A[9], const float B[9], float C[9]) {
#pragma unroll
    for (int r = 0; r < 3; ++r)
#pragma unroll
        for (int c = 0; c < 3; ++c)
            C[r*3+c] = A[r*3+0]*B[0*3+c] + A[r*3+1]*B[1*3+c] + A[r*3+2]*B[2*3+c];
}

// cos( rel_y[i], G_parent @ template[i] ) with the reference's eps handling
__device__ __forceinline__ float jointCos(const float G[9], const float* __restrict__ yr,
                                          int i, int pa) {
    float tx = yr[45 + 3*i + 0], ty = yr[45 + 3*i + 1], tz = yr[45 + 3*i + 2];
    float px = G[0]*tx + G[1]*ty + G[2]*tz;
    float py = G[3]*tx + G[4]*ty + G[5]*tz;
    float pz = G[6]*tx + G[7]*ty + G[8]*tz;
    float ryx = yr[3*i + 0] - yr[3*pa + 0];
    float ryy = yr[3*i + 1] - yr[3*pa + 1];
    float ryz = yr[3*i + 2] - yr[3*pa + 2];
    float dot = ryx*px + ryy*py + ryz*pz;
    float na = fmaxf(sqrtf(ryx*ryx + ryy*ryy + ryz*ryz), 1e-8f);
    float nb = fmaxf(sqrtf(px*px + py*py + pz*pz), 1e-8f);
    return dot / (na * nb);
}

// ---------- kernel A: per-element compute + WMMA block reduction ----------

__global__ __launch_bounds__(TPB)
void fk_partial(const float* __restrict__ pred, const float* __restrict__ y,
                float* __restrict__ partial, int ntiles) {
    // staging: 64 pred rows (5760 f) + 64 y rows (6336 f) = 12096 floats = 48384 B
    __shared__ float lds[TPB*PRED_F + TPB*Y_F];
    const int tid = threadIdx.x;

    float acc[16];
#pragma unroll
    for (int j = 0; j < 16; ++j) acc[j] = 0.0f;

    for (int tile = blockIdx.x; tile < ntiles; tile += gridDim.x) {
        __syncthreads();
        // coalesced, 16B-aligned staging (tile bases: 23040B / 25344B multiples)
        const float4* gp = (const float4*)(pred + (size_t)tile * (TPB*PRED_F));
        float4* lp = (float4*)lds;
        for (int i = tid; i < TPB*PRED_F/4; i += TPB) lp[i] = gp[i];
        const float4* gy = (const float4*)(y + (size_t)tile * (TPB*Y_F));
        float4* ly = (float4*)(lds + TPB*PRED_F);
        for (int i = tid; i < TPB*Y_F/4; i += TPB) ly[i] = gy[i];
        __syncthreads();

        const float* pr = lds + tid*PRED_F;
        const float* yr = lds + TPB*PRED_F + tid*Y_F;

        float Rr[9];
#pragma unroll
        for (int k = 0; k < 9; ++k) Rr[k] = yr[90 + k];

        float L[9], G0[9], G1[9];
        gs6(pr + 0, L);  mm3(Rr, L, G0);          // G0 = R_root @ L0
        acc[1]  += jointCos(G0, yr, 1, 0);        // rel_pred[1]  = G0 @ t1
        acc[14] += jointCos(G0, yr, 14, 0);       // rel_pred[14] = G0 @ t14
        gs6(pr + 6, L);  mm3(G0, L, G1);          // G1 = G0 @ L1

        const int chainStart[4] = {2, 5, 8, 11};
#pragma unroll
        for (int c = 0; c < 4; ++c) {
            int s = chainStart[c];
            float Ga[9], Gb[9];
            acc[s]   += jointCos(G1, yr, s, 1);       // parent of s is joint 1
            gs6(pr + 6*s, L);       mm3(G1, L, Ga);   // G[s]
            acc[s+1] += jointCos(Ga, yr, s+1, s);
            gs6(pr + 6*(s+1), L);   mm3(Ga, L, Gb);   // G[s+1]
            acc[s+2] += jointCos(Gb, yr, s+2, s+1);
            // s+2 is a leaf: its local rotation is never needed
        }
        // acc[0] stays 0 (rel vectors are both zero -> cos = 0); acc[15] pad = 0
    }

    // ---- block reduction via V_WMMA_F32_16X16X4_F32: D = S_chunk(16x4) x ones(4x16) + C ----
    __syncthreads();
    float* S = lds;                       // S[64][16]
#pragma unroll
    for (int j = 0; j < 16; ++j) S[tid*16 + j] = acc[j];
    __syncthreads();

    const int lane = tid & 31;
    const int w    = tid >> 5;            // wave id (0..1), each reduces 32 rows
    const int m    = lane & 15;           // A-matrix M index (doc: 16x4 f32 A layout)
    const int kb   = (lane < 16) ? 0 : 2; // VGPR0 holds K=0/2, VGPR1 holds K=1/3

    v8f dacc = {};
    v2f ones; ones.x = 1.0f; ones.y = 1.0f;   // all-ones B is layout-invariant
#pragma unroll
    for (int c = 0; c < 8; ++c) {
        int r0 = w*32 + c*4 + kb;
        v2f a;
        a.x = S[(r0    )*16 + m];   // A[m][kb]   = acc_m of row r0
        a.y = S[(r0 + 1)*16 + m];   // A[m][kb+1] = acc_m of row r0+1
        dacc = __builtin_amdgcn_wmma_f32_16x16x4_f32(
            false, a, false, ones, (short)0, dacc, false, false);
    }
    // D[m][n] = sum over this wave's 32 rows of acc_m (same for every n); take column N=0
    float* W = lds + 1024;                // W[2][16]
    if ((lane & 15) == 0) {
        int mb = (lane >= 16) ? 8 : 0;    // lanes 0/16 hold M=0..7 / M=8..15 in v0..v7
#pragma unroll
        for (int i = 0; i < 8; ++i) W[w*16 + mb + i] = dacc[i];
    }
    __syncthreads();
    if (tid < 16)
        partial[(size_t)blockIdx.x * 16 + tid] = W[tid] + W[16 + tid];
}

// ---------- kernel B: deterministic final reduction ----------

__global__ __launch_bounds__(256)
void fk_final(const float* __restrict__ partial, int nblocks, float invB,
              float* __restrict__ out) {
    __shared__ float P[16*16];
    int tid = threadIdx.x;
    int j   = tid & 15;
    int seg = tid >> 4;                   // 16 segments
    float s = 0.0f;
    for (int b = seg; b < nblocks; b += 16) s += partial[(size_t)b*16 + j];
    P[seg*16 + j] = s;
    __syncthreads();
    if (tid < 15) {
        float t = 0.0f;
#pragma unroll
        for (int g = 0; g < 16; ++g) t += P[g*16 + tid];
        out[tid] = 50.0f - 50.0f * invB * t;   // 50*mean(1-cos) = 50 - 50*sum(cos)/B
    }
}

extern "C" void kernel_launch(void* const* d_in, const int* in_sizes, int n_in,
                              void* d_out, int out_size, void* d_ws, size_t ws_size,
                              hipStream_t stream) {
    const float* pred = (const float*)d_in[0];
    const float* y    = (const float*)d_in[1];
    int B      = in_sizes[0] / PRED_F;        // 524288
    int ntiles = B / TPB;                     // 8192
    int grid   = (ntiles < GRID_A) ? ntiles : GRID_A;
    float* partial = (float*)d_ws;            // grid*16 floats

    fk_partial<<<grid, TPB, 0, stream>>>(pred, y, partial, ntiles);
    fk_final<<<1, 256, 0, stream>>>(partial, grid, 1.0f / (float)B, (float*)d_out);
}